// M_GAT_orig_48249662603676
// MI455X (gfx1250) — compile-verified
//
#include <hip/hip_runtime.h>
#include <hip/hip_bf16.h>

// ---------------------------------------------------------------------------
// Heterogeneous 3-layer GAT (DGL GATConv semantics) for MI455X / gfx1250.
// GEMMs use V_WMMA_F32_16X16X4_F32 (fp32, matches reference precision) with a
// pre-packed B layout so the inner loop is branch-free: two global_load_b64
// + one v_wmma per K-step. Edge softmax uses uint-keyed atomicMax + atomicAdd
// (L2-resident: each N x 192 fp32 buffer is 38.4 MB < 192 MB L2).
// ---------------------------------------------------------------------------

typedef __attribute__((ext_vector_type(2))) float v2f;
typedef __attribute__((ext_vector_type(8))) float v8f;

#define GAT_N  50000
#define GAT_E  250000
#define GAT_NT 3

// ---- helpers ---------------------------------------------------------------

__device__ __forceinline__ unsigned f2key(float f) {
    unsigned u = __float_as_uint(f);
    return (u & 0x80000000u) ? ~u : (u | 0x80000000u);
}
__device__ __forceinline__ float key2f(unsigned k) {
    unsigned u = (k & 0x80000000u) ? (k & 0x7FFFFFFFu) : ~k;
    return __uint_as_float(u);
}

__global__ void fill_u32(unsigned* __restrict__ p, unsigned v, long n) {
    long i = (long)blockIdx.x * blockDim.x + threadIdx.x;
    if (i < n) p[i] = v;
}

// ---- feature assembly: x0[n, 0:64] = [emb0|emb1|emb2|cont|zeros] -----------

__global__ void build_features(const int* __restrict__ cat,
                               const float* __restrict__ cont,
                               const float* __restrict__ e0,
                               const float* __restrict__ e1,
                               const float* __restrict__ e2,
                               float* __restrict__ x0) {
    long i = (long)blockIdx.x * blockDim.x + threadIdx.x;
    if (i >= (long)GAT_N * 64) return;
    int n = (int)(i >> 6), c = (int)(i & 63);
    float v = 0.f;
    if (c < 8)       v = e0[cat[n * 3 + 0] * 8 + c];
    else if (c < 11) v = e1[cat[n * 3 + 1] * 3 + (c - 8)];
    else if (c < 17) v = e2[cat[n * 3 + 2] * 6 + (c - 11)];
    else if (c < 34) v = cont[n * 17 + (c - 17)];
    x0[i] = v;
}

// ---- pack W (Kreal x M, zero-pad to Kloop) into Wb[k/2][col][2] ------------
// Lane fragment (W[ka][col], W[ka+1][col]) becomes one aligned b64 load.

__global__ void pack_w(const float* __restrict__ W, int Kreal, int Kloop, int M,
                       float* __restrict__ Wb) {
    int i = blockIdx.x * blockDim.x + threadIdx.x;
    if (i >= Kloop * M) return;
    int k = i / M, col = i - k * M;
    float v = (k < Kreal) ? W[(size_t)k * M + col] : 0.f;
    Wb[(((size_t)(k >> 1) * M + col) << 1) | (unsigned)(k & 1)] = v;
}

// ---- fp32 WMMA GEMM: C(N x M) = X(N x K, stride ldx) @ W(K x M) ------------
// One wave32 per 16x16 output tile; K stepped by 4 via V_WMMA_F32_16X16X4_F32.
// A layout (ISA 7.12.2): lanes 0-15 hold K=k0,k0+1; lanes 16-31 hold k0+2,k0+3.
// B pre-packed (pack_w): pair p = k0/2 + half at Wb + (p*M + col)*2.
// C layout: VGPR i -> row (i | half*8), col = lane & 15.

__global__ __launch_bounds__(32)
void gemm_wmma_f32(const float* __restrict__ X, int ldx,
                   const float* __restrict__ Wb, int Kloop,
                   int M, float* __restrict__ C) {
    const int lane = threadIdx.x;
    const int r    = lane & 15;
    const int half = lane >> 4;
    const int row0 = blockIdx.x * 16;
    const int col  = blockIdx.y * 16 + r;
    const float* __restrict__ Xrow = X + (size_t)(row0 + r) * ldx + 2 * half;
    const float* __restrict__ Wp   = Wb + ((size_t)half * M + col) * 2;
    v8f acc = {};
    for (int k0 = 0; k0 < Kloop; k0 += 4) {
        __builtin_prefetch(Xrow + k0 + 8, 0, 1);          // -> global_prefetch_b8
        v2f a = *(const v2f*)(Xrow + k0);                  // X zero-padded past Kreal
        v2f b = *(const v2f*)(Wp + (size_t)(k0 >> 1) * (2 * M));
        acc = __builtin_amdgcn_wmma_f32_16x16x4_f32(
            /*neg_a=*/false, a, /*neg_b=*/false, b,
            /*c_mod=*/(short)0, acc, /*reuse_a=*/false, /*reuse_b=*/false);
    }
#pragma unroll
    for (int i = 0; i < 8; ++i) {
        int row = row0 + (half ? (8 + i) : i);
        C[(size_t)row * M + col] = acc[i];
    }
}

// ---- el/er: per (node, head) reductions ------------------------------------

__global__ void attn_coef(const float* __restrict__ feat,
                          const float* __restrict__ al,
                          const float* __restrict__ ar,
                          int NH, int H, int D,
                          float* __restrict__ el, float* __restrict__ er) {
    int i = blockIdx.x * blockDim.x + threadIdx.x;  // i = n*H + h
    if (i >= NH) return;
    const float* f  = feat + (size_t)i * D;
    const float* a1 = al + (size_t)(i % H) * D;
    const float* a2 = ar + (size_t)(i % H) * D;
    float s1 = 0.f, s2 = 0.f;
    for (int d = 0; d < D; ++d) { float v = f[d]; s1 += v * a1[d]; s2 += v * a2[d]; }
    el[i] = s1;
    er[i] = s2;
}

// ---- edge pass 1: logits + segment max (uint-keyed float atomicMax) --------

__global__ void edge_logits(const int* __restrict__ src, const int* __restrict__ dst,
                            const float* __restrict__ el, const float* __restrict__ er,
                            int H, float* __restrict__ ebuf,
                            unsigned* __restrict__ mkey) {
    long i = (long)blockIdx.x * blockDim.x + threadIdx.x;
    if (i >= (long)GAT_E * H) return;
    int e = (int)(i / H), h = (int)(i - (long)e * H);
    int s = src[e], d = dst[e];
    float v = el[(size_t)s * H + h] + er[(size_t)d * H + h];
    v = (v > 0.f) ? v : 0.2f * v;                  // LeakyReLU(0.2)
    ebuf[i] = v;
    atomicMax(&mkey[(size_t)d * H + h], f2key(v));
}

// ---- edge pass 2: exp(e - max) + segment sum -------------------------------

__global__ void edge_expsum(const int* __restrict__ dst,
                            const unsigned* __restrict__ mkey,
                            int H, float* __restrict__ ebuf,
                            float* __restrict__ den) {
    long i = (long)blockIdx.x * blockDim.x + threadIdx.x;
    if (i >= (long)GAT_E * H) return;
    int e = (int)(i / H), h = (int)(i - (long)e * H);
    int d = dst[e];
    float m  = key2f(mkey[(size_t)d * H + h]);
    float ee = __expf(ebuf[i] - m);
    ebuf[i] = ee;
    atomicAdd(&den[(size_t)d * H + h], ee);
}

// ---- edge pass 3: alpha-weighted message scatter ---------------------------

__global__ void edge_scatter(const int* __restrict__ src, const int* __restrict__ dst,
                             const float* __restrict__ ebuf, const float* __restrict__ den,
                             const float* __restrict__ feat,
                             int H, int D, long total, float* __restrict__ acc) {
    const int F = H * D;
    for (long i = (long)blockIdx.x * blockDim.x + threadIdx.x; i < total;
         i += (long)gridDim.x * blockDim.x) {
        long e = i / F;
        int hd = (int)(i - e * F);
        int h  = hd / D;
        int s = src[e], d = dst[e];
        float alpha = ebuf[(size_t)e * H + h] / den[(size_t)d * H + h];
        atomicAdd(&acc[(size_t)d * F + hd], feat[(size_t)s * F + hd] * alpha);
    }
}

// ---- finalize: agg += (1/NT) * leaky_relu(acc + bias, 0.01) ----------------

__global__ void finalize_type(const float* __restrict__ acc,
                              const float* __restrict__ bias,
                              long NF, int F, float invNT,
                              float* __restrict__ agg) {
    long i = (long)blockIdx.x * blockDim.x + threadIdx.x;
    if (i >= NF) return;
    float v = acc[i] + bias[i % F];
    v = (v > 0.f) ? v : 0.01f * v;
    agg[i] += invNT * v;
}

// ---------------------------------------------------------------------------

static inline int cdiv_i(long a, long b) { return (int)((a + b - 1) / b); }

extern "C" void kernel_launch(void* const* d_in, const int* in_sizes, int n_in,
                              void* d_out, int out_size, void* d_ws, size_t ws_size,
                              hipStream_t stream) {
    (void)in_sizes; (void)n_in; (void)out_size; (void)ws_size;
    const int N = GAT_N, E = GAT_E, NT = GAT_NT;
    const int TB = 256;

    // setup_inputs() dict order
    const int*   cat  = (const int*)d_in[0];
    const float* cont = (const float*)d_in[1];
    const int*   srcA = (const int*)d_in[2];   // (NT, E)
    const int*   dstA = (const int*)d_in[3];   // (NT, E)
    const float* e0   = (const float*)d_in[4];
    const float* e1   = (const float*)d_in[5];
    const float* e2   = (const float*)d_in[6];
    const float* Wl[3]  = {(const float*)d_in[7],  (const float*)d_in[11], (const float*)d_in[15]};
    const float* all[3] = {(const float*)d_in[8],  (const float*)d_in[12], (const float*)d_in[16]};
    const float* arl[3] = {(const float*)d_in[9],  (const float*)d_in[13], (const float*)d_in[17]};
    const float* bl[3]  = {(const float*)d_in[10], (const float*)d_in[14], (const float*)d_in[18]};

    // workspace carve-out (256B aligned)
    char* base = (char*)d_ws;
    auto carve = [&](size_t bytes) -> void* {
        void* p = (void*)base;
        base += (bytes + 255) & ~(size_t)255;
        return p;
    };
    float*    x0   = (float*)carve((size_t)N * 64 * 4);    // padded layer-1 input
    float*    bufA = (float*)carve((size_t)N * 192 * 4);   // layer-1 output
    float*    bufB = (float*)carve((size_t)N * 192 * 4);   // layer-2 output
    float*    feat = (float*)carve((size_t)N * 192 * 4);   // per-type linear feat
    float*    acc  = (float*)carve((size_t)N * 192 * 4);   // per-type message sum
    float*    el   = (float*)carve((size_t)N * 3 * 4);
    float*    er   = (float*)carve((size_t)N * 3 * 4);
    float*    den  = (float*)carve((size_t)N * 3 * 4);
    unsigned* mkey = (unsigned*)carve((size_t)N * 3 * 4);
    float*    ebuf = (float*)carve((size_t)E * 3 * 4);
    float*    Wb   = (float*)carve((size_t)192 * 192 * 4); // packed per-type weights

    // 1) assemble node features (N x 64, zero padded past col 34)
    build_features<<<cdiv_i((long)N * 64, TB), TB, 0, stream>>>(cat, cont, e0, e1, e2, x0);

    // layer configs
    const int   Kreal_[3] = {34, 192, 192};
    const int   Kloop_[3] = {36, 192, 192};
    const int   ldx_[3]   = {64, 192, 192};
    const int   Hh_[3]    = {3, 3, 1};
    const int   Dd_[3]    = {64, 64, 32};
    const float* Xin_[3]  = {x0, bufA, bufB};
    float*       Agg_[3]  = {bufA, bufB, (float*)d_out};
    // key of -inf: bits(-inf)=0xFF800000, sign set => key = ~bits = 0x007FFFFF
    const unsigned NEG_INF_KEY = 0x007FFFFFu;

    for (int L = 0; L < 3; ++L) {
        const int H = Hh_[L], D = Dd_[L], F = H * D;
        const int Kreal = Kreal_[L], Kloop = Kloop_[L], ldx = ldx_[L];
        const float* X = Xin_[L];
        float* agg = Agg_[L];
        const long NF = (long)N * F;
        const long NH = (long)N * H;
        const long EH = (long)E * H;
        const long EF = (long)E * F;

        fill_u32<<<cdiv_i(NF, TB), TB, 0, stream>>>((unsigned*)agg, 0u, NF);

        for (int t = 0; t < NT; ++t) {
            const float* Wt = Wl[L] + (size_t)t * Kreal * F;
            const float* at = all[L] + (size_t)t * F;   // (H, D) contiguous
            const float* rt = arl[L] + (size_t)t * F;
            const float* bt = bl[L] + (size_t)t * F;
            const int* src_t = srcA + (size_t)t * E;
            const int* dst_t = dstA + (size_t)t * E;

            // pack weights for branch-free WMMA B-fragment loads
            pack_w<<<cdiv_i((long)Kloop * F, TB), TB, 0, stream>>>(Wt, Kreal, Kloop, F, Wb);

            // feat = X @ W[t]   (wave32 WMMA tiles; N = 3125 * 16 exactly)
            dim3 ggrid(N / 16, F / 16);
            gemm_wmma_f32<<<ggrid, 32, 0, stream>>>(X, ldx, Wb, Kloop, F, feat);

            // per-(node, head) attention coefficients
            attn_coef<<<cdiv_i(NH, TB), TB, 0, stream>>>(feat, at, rt, (int)NH, H, D, el, er);

            // reset segment buffers
            fill_u32<<<cdiv_i(NH, TB), TB, 0, stream>>>(mkey, NEG_INF_KEY, NH);
            fill_u32<<<cdiv_i(NH, TB), TB, 0, stream>>>((unsigned*)den, 0u, NH);
            fill_u32<<<cdiv_i(NF, TB), TB, 0, stream>>>((unsigned*)acc, 0u, NF);

            // edge softmax + scatter
            edge_logits<<<cdiv_i(EH, TB), TB, 0, stream>>>(src_t, dst_t, el, er, H, ebuf, mkey);
            edge_expsum<<<cdiv_i(EH, TB), TB, 0, stream>>>(dst_t, mkey, H, ebuf, den);
            edge_scatter<<<cdiv_i(EF, TB), TB, 0, stream>>>(src_t, dst_t, ebuf, den, feat,
                                                            H, D, EF, acc);

            // agg += (1/NT) * leaky_relu(acc + b[t], 0.01)
            finalize_type<<<cdiv_i(NF, TB), TB, 0, stream>>>(acc, bt, NF, F, 1.0f / NT, agg);
        }
    }
}